// Mask2FormerLoss_62577673502905
// MI455X (gfx1250) — compile-verified
//
#include <hip/hip_runtime.h>
#include <math.h>

// ---------------- problem constants (from reference) ----------------
#define NCLS    40
#define CP1     41          // NUM_CLASSES + 1
#define B_      4
#define Q_      100
#define H_      256
#define W_      256
#define P_      (H_ * W_)   // 65536 pixels per batch
#define IGNORE_ 255
#define EOS_W   0.1f
#define EPS_    1e-8f

// WMMA tiling: Q padded to 7 M-tiles, classes padded to 3 N-tiles.
// Column 40 of the padded one-hot is identically 1.0 -> src_sum for free.
#define QPAD    112
#define CPAD    48
#define STAGE   64                     // pixels per TDM stage (K chunk pool)
#define PIX_PER_WG 4096
#define NSTAGES (PIX_PER_WG / STAGE)   // 64
#define LDS_STRIDE 65                  // 64 + 1 pad dword (TDM pad fields)

typedef __attribute__((ext_vector_type(2))) float v2f;
typedef __attribute__((ext_vector_type(8))) float v8f;
typedef __attribute__((ext_vector_type(4))) unsigned int u32x4;
typedef __attribute__((ext_vector_type(4))) int i32x4;
typedef __attribute__((ext_vector_type(8))) int i32x8;

// ---------------- workspace layout (floats) ----------------
#define WS_COUNTS 0                               // [B][Q][NCLS]   = 16000
#define WS_TSUM   (WS_COUNTS + B_*Q_*NCLS)        // [B][NCLS]      = 160
#define WS_CESUM  (WS_TSUM + B_*NCLS)             // scalar
#define WS_NVALID (WS_CESUM + 1)                  // scalar
#define WS_INTER  (WS_NVALID + 31)                // [B][QPAD][CPAD] (aligned)
#define WS_TOTAL  (WS_INTER + B_*QPAD*CPAD)

// fast sigmoid: v_rcp_f32 instead of IEEE division (feeds a 65k-element
// reduction; ~1ulp rcp is plenty and saves ~10 VALU per element)
__device__ __forceinline__ float fast_sigmoid(float x) {
    return __builtin_amdgcn_rcpf(1.0f + __expf(-x));
}

// ============================================================
// Kernel 0: zero the accumulator workspace
// ============================================================
__global__ void zero_ws_kernel(float* __restrict__ ws, int n) {
    int i = blockIdx.x * blockDim.x + threadIdx.x;
    if (i < n) ws[i] = 0.0f;
}

// ============================================================
// Kernel 1: fused per-pixel pass over pred_masks (one 105MB sweep)
//   - online logsumexp over Q + argmax over Q in a single loop
//   - captures logit at q == tgt for the pixel CE
//   - histograms counts[b, argmax_q, tgt] and tsum[b, tgt]
// ============================================================
__global__ __launch_bounds__(256) void pixel_pass_kernel(
    const float* __restrict__ pm, const int* __restrict__ tg,
    float* __restrict__ ws) {
    const int gid = blockIdx.x * blockDim.x + threadIdx.x;   // [0, B*P)
    const int b = gid / P_;
    const int p = gid - b * P_;

    const int t = tg[gid];
    const bool valid = (t != IGNORE_);
    const int ts = valid ? t : 0;   // safe index (< Q_)

    const float* base = pm + (size_t)b * Q_ * P_ + p;

    float m = -3.4e38f, s = 0.0f, xt = 0.0f;
    float best = -3.4e38f;
    int amax = 0;
    #pragma unroll 4
    for (int q = 0; q < Q_; ++q) {
        const float x = base[(size_t)q * P_];        // coalesced across lanes
        if (x > best) { best = x; amax = q; }        // first-max tie break
        const float mn = fmaxf(m, x);                // streaming logsumexp
        s = s * __expf(m - mn) + __expf(x - mn);
        m = mn;
        if (q == ts) xt = x;
    }
    const float nll = (m + __logf(s)) - xt;

    __shared__ float red[256];
    red[threadIdx.x] = valid ? nll : 0.0f;
    __syncthreads();
    for (int o = 128; o > 0; o >>= 1) {
        if (threadIdx.x < o) red[threadIdx.x] += red[threadIdx.x + o];
        __syncthreads();
    }
    if (threadIdx.x == 0) atomicAdd(&ws[WS_CESUM], red[0]);
    __syncthreads();
    red[threadIdx.x] = valid ? 1.0f : 0.0f;
    __syncthreads();
    for (int o = 128; o > 0; o >>= 1) {
        if (threadIdx.x < o) red[threadIdx.x] += red[threadIdx.x + o];
        __syncthreads();
    }
    if (threadIdx.x == 0) atomicAdd(&ws[WS_NVALID], red[0]);

    if (valid) {
        atomicAdd(&ws[WS_COUNTS + (b * Q_ + amax) * NCLS + t], 1.0f);
        atomicAdd(&ws[WS_TSUM + b * NCLS + t], 1.0f);
    }
}

// ============================================================
// TDM helper: DMA one 100-row x 64-col f32 tile (row stride P_ elements)
// from global into LDS, inserting 1 pad dword after every 64 dwords so
// rows land at stride 65 (bank-conflict-free for the WMMA A reads).
// D# per CDNA5 ISA ch.8: group0 = {flags, lds_addr, global_addr, type},
// group1 = {data_size/pad, tensor dims, tile dims, strides}; groups 2/3
// NULL-equivalent (2D tensor). Tracked by TENSORcnt.
// ============================================================
__device__ __forceinline__ void tdm_load_tile(const float* gptr, unsigned lds_addr) {
    const unsigned long long ga = (unsigned long long)(uintptr_t)gptr;
    u32x4 g0;
    g0[0] = 1u;                                        // count=1, user descriptor
    g0[1] = lds_addr;                                  // LDS byte address
    g0[2] = (unsigned)ga;                              // global_addr[31:0]
    g0[3] = (unsigned)((ga >> 32) & 0x1FFFFFFu)        // global_addr[56:32]
          | (2u << 30);                                // type = 2 ("image")
    i32x8 g1;
    g1[0] = (2 << 16)      // data_size = 4 bytes
          | (1 << 20)      // pad_enable
          | (5 << 22)      // pad_interval: 64 dwords
          | (0 << 25);     // pad_amount: 1 dword
    g1[1] = (int)((P_ & 0xFFFF) << 16);                // tensor_dim0[15:0]   (=0)
    g1[2] = (int)((P_ >> 16) | (Q_ << 16));            // dim0[31:16]=1, dim1=100
    g1[3] = (int)(STAGE << 16);                        // dim1 hi=0, tile_dim0=64
    g1[4] = Q_;                                        // tile_dim1=100, tile_dim2=0
    g1[5] = P_;                                        // tensor_dim0_stride lo32
    g1[6] = 0;                                         // stride hi16, dim1_stride lo
    g1[7] = 0;
    const i32x4 z4 = {0, 0, 0, 0};
    const i32x8 z8 = {0, 0, 0, 0, 0, 0, 0, 0};
    // clang-23 / therock-10.0 form: (g0, g1, g2, g3, g4, cpol)
    __builtin_amdgcn_tensor_load_to_lds(g0, g1, z4, z4, z8, 0);
}

// ============================================================
// Kernel 2: dice contraction on the f32 matrix pipe.
//   inter_ext[b, q, c] = sum_p sigmoid(pm[b,q,p]) * onehot(tgt[b,p], c)
//   with column NCLS(=40) forced to 1.0 => src_sum[b,q] computed for free.
//   Double-buffered TDM staging: wave 0 issues tensor_load_to_lds for
//   stage s+1 while waves 0..6 compute WMMAs on stage s; wave 0 then
//   waits TENSORcnt==0 and the workgroup barrier publishes the buffer.
//   Waves 0..6 each own one 16-row M-tile of Q with 3 N-tile f32
//   accumulators; K loops over pixels in chunks of 4 via
//   v_wmma_f32_16x16x4_f32; B one-hot fragments synthesized in registers.
// ============================================================
__global__ __launch_bounds__(256) void wmma_inter_kernel(
    const float* __restrict__ pm, const int* __restrict__ tg,
    float* __restrict__ ws) {
    __shared__ float sA[2][QPAD * LDS_STRIDE];   // raw logits tile (double buffer)
    __shared__ int   sT[2][STAGE];               // targets (double buffer)

    const int chunks = P_ / PIX_PER_WG;          // 16
    const int b    = blockIdx.x / chunks;
    const int pix0 = (blockIdx.x - b * chunks) * PIX_PER_WG;

    const int tid  = threadIdx.x;
    const int lane = tid & 31;
    const int wave = tid >> 5;

    const float* pmb = pm + (size_t)b * Q_ * P_;
    const int*   tgb = tg + (size_t)b * P_;

    const unsigned ldsA0 = (unsigned)(uintptr_t)(&sA[0][0]);
    const unsigned ldsA1 = (unsigned)(uintptr_t)(&sA[1][0]);

    v8f acc0 = {0,0,0,0,0,0,0,0};
    v8f acc1 = {0,0,0,0,0,0,0,0};
    v8f acc2 = {0,0,0,0,0,0,0,0};

    // Per-lane fragment coordinates (wave32 WMMA 16x16 layout):
    //   A (16x4): lanes 0-15 hold M=lane, VGPR0=K0/VGPR1=K1; lanes 16-31 K2/K3
    //   B (4x16): lanes hold N=lane&15; .x -> K row (lane<16?0:1), .y -> (lane<16?2:3)
    //   C/D (16x16): element j -> M = j + (lane>=16 ? 8 : 0), N = lane&15
    const int   mrow = wave * 16 + (lane & 15);
    const float rowv = (mrow < Q_) ? 1.0f : 0.0f;   // zero out the 12 pad rows
    const int   acol = (lane >> 4) * 2;             // 0 or 2
    const int   ncol = lane & 15;
    const int   kbx  = (lane < 16) ? 0 : 1;
    const int   kby  = (lane < 16) ? 2 : 3;

    // ---- prologue: DMA stage 0, load its targets ----
    if (wave == 0) tdm_load_tile(pmb + pix0, ldsA0);
    if (tid < STAGE) sT[0][tid] = tgb[pix0 + tid];
    if (wave == 0) __builtin_amdgcn_s_wait_tensorcnt(0);
    __syncthreads();

    for (int s = 0; s < NSTAGES; ++s) {
        const int cur = s & 1;
        const int nxt = cur ^ 1;

        // ---- kick off DMA + target load for the next stage ----
        if (s + 1 < NSTAGES) {
            const int npix = pix0 + (s + 1) * STAGE;
            if (wave == 0) tdm_load_tile(pmb + npix, nxt ? ldsA1 : ldsA0);
            if (tid < STAGE) sT[nxt][tid] = tgb[npix + tid];
        }

        // ---- compute on current stage ----
        const float* sAc = &sA[cur][0];
        const int*   sTc = &sT[cur][0];
        if (wave < 7) {
            #pragma unroll 4
            for (int k0 = 0; k0 < STAGE; k0 += 4) {
                const float x0 = sAc[mrow * LDS_STRIDE + k0 + acol];
                const float x1 = sAc[mrow * LDS_STRIDE + k0 + acol + 1];
                v2f a;
                a.x = rowv * fast_sigmoid(x0);   // v_rcp_f32 path, pad rows -> 0
                a.y = rowv * fast_sigmoid(x1);

                const int tx = sTc[k0 + kbx];
                const int ty = sTc[k0 + kby];

                // N-tile 0: classes 0..15
                v2f b0;
                b0.x = (tx == ncol) ? 1.0f : 0.0f;
                b0.y = (ty == ncol) ? 1.0f : 0.0f;
                acc0 = __builtin_amdgcn_wmma_f32_16x16x4_f32(
                    false, a, false, b0, (short)0, acc0, false, false);

                // N-tile 1: classes 16..31
                const int c1 = 16 + ncol;
                v2f b1;
                b1.x = (tx == c1) ? 1.0f : 0.0f;
                b1.y = (ty == c1) ? 1.0f : 0.0f;
                acc1 = __builtin_amdgcn_wmma_f32_16x16x4_f32(
                    false, a, false, b1, (short)0, acc1, false, false);

                // N-tile 2: classes 32..39, col 40 == all-ones (src_sum), 41..47 zero
                const int c2 = 32 + ncol;
                v2f b2;
                b2.x = (c2 == NCLS) ? 1.0f : ((tx == c2) ? 1.0f : 0.0f);
                b2.y = (c2 == NCLS) ? 1.0f : ((ty == c2) ? 1.0f : 0.0f);
                acc2 = __builtin_amdgcn_wmma_f32_16x16x4_f32(
                    false, a, false, b2, (short)0, acc2, false, false);
            }
        }

        // ---- publish next buffer ----
        if (wave == 0) __builtin_amdgcn_s_wait_tensorcnt(0);
        __syncthreads();
    }

    if (wave < 7) {
        float* inter = ws + WS_INTER + b * QPAD * CPAD;
        const int moff = (lane >= 16) ? 8 : 0;
        #pragma unroll
        for (int j = 0; j < 8; ++j) {
            const int q = wave * 16 + j + moff;
            atomicAdd(&inter[q * CPAD + ncol],      acc0[j]);
            atomicAdd(&inter[q * CPAD + 16 + ncol], acc1[j]);
            atomicAdd(&inter[q * CPAD + 32 + ncol], acc2[j]);
        }
    }
}

// ============================================================
// Kernel 3: finalize (single block)
// ============================================================
__global__ __launch_bounds__(256) void finalize_kernel(
    const float* __restrict__ logits, const float* __restrict__ ws,
    float* __restrict__ out) {
    __shared__ float red[256];
    __shared__ float dice_sum[NCLS];
    const int tid = threadIdx.x;

    // ---- loss_ce: 400 (b,q) terms ----
    float ce_part = 0.0f;
    for (int i = tid; i < B_ * Q_; i += 256) {
        const float* cnt = ws + WS_COUNTS + i * NCLS;
        float tot = 0.0f, bestv = -1.0f;
        int bestc = 0;
        for (int c = 0; c < NCLS; ++c) {
            const float v = cnt[c];
            tot += v;
            if (v > bestv) { bestv = v; bestc = c; }   // ties -> smallest class
        }
        const int tc = (tot > 0.0f) ? bestc : NCLS;

        const float* lg = logits + i * CP1;
        float mx = lg[0];
        for (int c = 1; c < CP1; ++c) mx = fmaxf(mx, lg[c]);
        float se = 0.0f;
        for (int c = 0; c < CP1; ++c) se += __expf(lg[c] - mx);
        const float lse = mx + __logf(se);
        const float nll = lse - lg[tc];
        const float w = (tc == NCLS) ? EOS_W : 1.0f;
        const float ce = (tc != NCLS) ? nll * w : 0.0f;  // ignore_index == NCLS
        const float pr = __expf(-ce);
        ce_part += (1.0f - pr) * (1.0f - pr) * ce;       // focal modulation
    }
    red[tid] = ce_part;
    __syncthreads();
    for (int o = 128; o > 0; o >>= 1) {
        if (tid < o) red[tid] += red[tid + o];
        __syncthreads();
    }
    const float loss_ce = red[0] / (float)(B_ * Q_);
    __syncthreads();

    // ---- dice ----
    if (tid < NCLS) dice_sum[tid] = 0.0f;
    __syncthreads();
    const float* inter = ws + WS_INTER;
    for (int i = tid; i < B_ * Q_ * NCLS; i += 256) {
        const int c  = i % NCLS;
        const int bq = i / NCLS;
        const int b  = bq / Q_;
        const int q  = bq - b * Q_;
        const float* row = inter + (b * QPAD + q) * CPAD;
        const float num  = row[c];
        const float ssum = row[NCLS];                      // ones-column = src_sum
        const float ts   = ws[WS_TSUM + b * NCLS + c];
        atomicAdd(&dice_sum[c], 2.0f * num / (ssum + ts + EPS_));
    }
    __syncthreads();

    float dl_part = 0.0f;
    if (tid < NCLS) {
        float tsc = 0.0f;
        for (int b = 0; b < B_; ++b) tsc += ws[WS_TSUM + b * NCLS + tid];
        const float mean = dice_sum[tid] / (float)(B_ * Q_);
        dl_part = (tsc > 0.0f) ? (1.0f - mean) : 0.0f;     // present classes only
    }
    red[tid] = dl_part;
    __syncthreads();
    for (int o = 128; o > 0; o >>= 1) {
        if (tid < o) red[tid] += red[tid + o];
        __syncthreads();
    }
    if (tid == 0) {
        const float dice_loss = red[0] / (float)NCLS;
        const float ce_mask = ws[WS_CESUM] / fmaxf(ws[WS_NVALID], 1.0f);
        out[0] = 2.0f * loss_ce + 5.0f * ce_mask + 5.0f * dice_loss;
    }
}

// ============================================================
extern "C" void kernel_launch(void* const* d_in, const int* in_sizes, int n_in,
                              void* d_out, int out_size, void* d_ws, size_t ws_size,
                              hipStream_t stream) {
    const float* pred_logits = (const float*)d_in[0];   // [B,Q,41]
    const float* pred_masks  = (const float*)d_in[1];   // [B,Q,H,W]
    const int*   targets     = (const int*)d_in[2];     // [B,H,W]
    float* ws  = (float*)d_ws;
    float* out = (float*)d_out;

    zero_ws_kernel<<<(WS_TOTAL + 255) / 256, 256, 0, stream>>>(ws, WS_TOTAL);
    pixel_pass_kernel<<<(B_ * P_) / 256, 256, 0, stream>>>(pred_masks, targets, ws);
    wmma_inter_kernel<<<B_ * (P_ / PIX_PER_WG), 256, 0, stream>>>(pred_masks, targets, ws);
    finalize_kernel<<<1, 256, 0, stream>>>(pred_logits, ws, out);
}